// CenterLoss_17884243820955
// MI455X (gfx1250) — compile-verified
//
#include <hip/hip_runtime.h>

// CenterLoss collapsed form:
//   loss = (1/B) * sum_b clip(||e_b - c_{label_b}||^2, 1e-12, 1e12) + (C-1)*1e-12
// Per-row squared distance computed via V_WMMA_F32_16X16X4_F32 Gram diagonal
// of the difference tile (wave32, 16 rows per wave, K accumulated 4 at a time).

typedef __attribute__((ext_vector_type(2))) float v2f;
typedef __attribute__((ext_vector_type(8))) float v8f;

#define NUM_CLASSES 32000
#define FEAT_DIM    256
#define BATCH       4096
#define CLAMP_MIN   1e-12f
#define CLAMP_MAX   1e12f

// 32 blocks * 256 threads (8 waves). Each wave handles a 16-row tile.
__global__ __launch_bounds__(256) void center_loss_partial_kernel(
    const float* __restrict__ emb,      // [BATCH, FEAT_DIM]
    const int*   __restrict__ labels,   // [BATCH]
    const float* __restrict__ ctr,      // [NUM_CLASSES, FEAT_DIM]
    float*       __restrict__ partial)  // [gridDim.x]
{
    __shared__ float wave_sum[8];

    const int lane = threadIdx.x & 31;
    const int wave = threadIdx.x >> 5;
    const int rowBase = (blockIdx.x * 8 + wave) * 16;

    // WMMA f32 16x16x4 A-layout: lane L holds M = L&15, K = {2*(L>>4), 2*(L>>4)+1}
    // (B 4x16 assumed transpose-symmetric -> identical per-lane indexing with N = L&15).
    const int m    = lane & 15;
    const int half = lane >> 4;
    const int row  = rowBase + m;
    const int lbl  = labels[row];

    const float* e = emb + (size_t)row * FEAT_DIM + 2 * half;
    const float* c = ctr + (size_t)lbl * FEAT_DIM + 2 * half;

    v8f acc = {};
#pragma unroll 8
    for (int t = 0; t < FEAT_DIM / 4; ++t) {
        v2f ev = *(const v2f*)(e + 4 * t);
        v2f cv = *(const v2f*)(c + 4 * t);
        v2f d  = ev - cv;   // difference tile: Gram(d) diagonal = ||e-c||^2
        // D = A*B + C with A = B = diff tile
        acc = __builtin_amdgcn_wmma_f32_16x16x4_f32(
            /*neg_a=*/false, d, /*neg_b=*/false, d,
            /*c_mod=*/(short)0, acc, /*reuse_a=*/false, /*reuse_b=*/false);
    }

    // C/D layout: VGPR j: lanes 0-15 -> D[j][lane], lanes 16-31 -> D[j+8][lane-16].
    // Diagonal D[i][i]: i=0..7  -> lane i,    acc[i]
    //                   i=8..15 -> lane i+16, acc[i-8]
    float v = 0.0f;
#pragma unroll
    for (int j = 0; j < 8; ++j) {
        float dj = acc[j];
        v = (lane == j)      ? dj : v;
        v = (lane == j + 24) ? dj : v;
    }
    const bool hasDiag = (lane < 8) || (lane >= 24);
    // clip only real diagonal entries; masked-out zeros are handled analytically
    v = hasDiag ? fminf(fmaxf(v, CLAMP_MIN), CLAMP_MAX) : 0.0f;

    // wave32 butterfly reduction (fixed order -> deterministic)
#pragma unroll
    for (int off = 16; off > 0; off >>= 1)
        v += __shfl_xor(v, off, 32);

    if (lane == 0) wave_sum[wave] = v;
    __syncthreads();

    if (threadIdx.x == 0) {
        float s = 0.0f;
#pragma unroll
        for (int i = 0; i < 8; ++i) s += wave_sum[i];
        partial[blockIdx.x] = s;
    }
}

// Single wave: reduce 32 block partials, apply scale + analytic clip floor.
__global__ __launch_bounds__(32) void center_loss_final_kernel(
    const float* __restrict__ partial, float* __restrict__ out)
{
    const int lane = threadIdx.x & 31;
    float v = partial[lane];
#pragma unroll
    for (int off = 16; off > 0; off >>= 1)
        v += __shfl_xor(v, off, 32);
    if (lane == 0) {
        const float floor_term = (float)(NUM_CLASSES - 1) * 1e-12f; // masked zeros clipped up
        out[0] = v * (1.0f / (float)BATCH) + floor_term;            // LAMBDA_C == 1.0
    }
}

extern "C" void kernel_launch(void* const* d_in, const int* in_sizes, int n_in,
                              void* d_out, int out_size, void* d_ws, size_t ws_size,
                              hipStream_t stream) {
    (void)in_sizes; (void)n_in; (void)out_size; (void)ws_size;
    const float* emb    = (const float*)d_in[0];   // [4096, 256] f32
    const int*   labels = (const int*)d_in[1];     // [4096] int
    const float* ctr    = (const float*)d_in[2];   // [32000, 256] f32
    float* out     = (float*)d_out;
    float* partial = (float*)d_ws;                 // 32 floats of scratch

    center_loss_partial_kernel<<<32, 256, 0, stream>>>(emb, labels, ctr, partial);
    center_loss_final_kernel<<<1, 32, 0, stream>>>(partial, out);
}